// PolyPTE_85461259256168
// MI455X (gfx1250) — compile-verified
//
#include <hip/hip_runtime.h>
#include <hip/hip_bf16.h>

typedef __attribute__((ext_vector_type(2))) float v2f;
typedef __attribute__((ext_vector_type(8))) float v8f;

#define C_CTX 5
#define NNEG  20
#define D_DIM 128
#define NROW  (C_CTX + C_CTX * NNEG)   // 105 rows (5 pos + 100 neg)
#define NTILE ((NROW + 15) / 16)       // 7 tiles of 16 rows (pad to 112)

__device__ __forceinline__ float log_sigmoid(float x) {
    // numerically stable: min(x,0) - log1p(exp(-|x|))
    return fminf(x, 0.0f) - log1pf(__expf(-fabsf(x)));
}

__global__ __launch_bounds__(256) void sgns_wmma_kernel(
    const int*   __restrict__ iword,
    const int*   __restrict__ owords,
    const int*   __restrict__ nwords,
    const float* __restrict__ emb1,
    const float* __restrict__ emb2,
    float*       __restrict__ per_b,
    int Bn)
{
    __shared__ __align__(16) float v1s[8][D_DIM];   // one center vector per wave
    const int lane = threadIdx.x & 31;
    const int wave = threadIdx.x >> 5;
    int b = blockIdx.x * 8 + wave;
    if (b >= Bn) b = Bn - 1;   // duplicate work on tail, same value written

    // Cooperative load of center vector v1 = emb1[iword[b]] into LDS (32 x float4).
    {
        const long long ci = (long long)iword[b];
        const float4* src = (const float4*)(emb1 + ci * (long long)D_DIM);
        ((float4*)v1s[wave])[lane] = src[lane];
    }
    __syncthreads();

    const int m    = lane & 15;   // A-matrix row for this lane
    const int half = lane >> 4;   // 0 -> K pair {0,1}, 1 -> K pair {2,3}

    // Hoist B-matrix fragments for all 32 K-chunks: B[k,n] = v1[k] broadcast
    // over all 16 columns (same LDS address across each half-wave -> broadcast).
    v2f bfrag[32];
#pragma unroll
    for (int kc = 0; kc < 32; ++kc)
        bfrag[kc] = *(const v2f*)(&v1s[wave][4 * kc + 2 * half]);

    float acc = 0.0f;

    for (int tile = 0; tile < NTILE; ++tile) {
        // Row -> gathered embedding index (branch-free clamped gathers).
        const int row = tile * 16 + m;
        const int oc  = (row < C_CTX) ? row : (C_CTX - 1);
        int nc = row - C_CTX;
        nc = nc < 0 ? 0 : (nc > C_CTX * NNEG - 1 ? C_CTX * NNEG - 1 : nc);
        const long long idx = (row < C_CTX)
            ? (long long)owords[b * C_CTX + oc]
            : (long long)nwords[b * (C_CTX * NNEG) + nc];

        const float* arow = emb2 + idx * (long long)D_DIM + 2 * half;
        __builtin_prefetch(arow, 0, 1);   // global_prefetch_b8

        // D(16x16,f32) += A(16x4,f32) x B(4x16,f32), accumulated over K=128.
        v8f c = {};
#pragma unroll
        for (int kc = 0; kc < 32; ++kc) {
            v2f a = *(const v2f*)(arow + 4 * kc);
            c = __builtin_amdgcn_wmma_f32_16x16x4_f32(
                    false, a, false, bfrag[kc], (short)0, c, false, false);
        }

        // D columns are all identical (B replicated over n). C/D layout:
        // VGPR j holds row j (lanes 0-15) and row j+8 (lanes 16-31).
        // Lanes 0-7 take row tile*16+j, lanes 16-23 take row tile*16+8+j.
        const int j = lane & 7;
        float s = c[0];
#pragma unroll
        for (int t = 1; t < 8; ++t)
            if (j == t) s = c[t];

        if ((lane & 8) == 0) {
            const int myrow = tile * 16 + half * 8 + j;
            if (myrow < NROW) {
                const float x = (myrow < C_CTX) ? s : -s;  // negatives flipped
                acc += log_sigmoid(x);
            }
        }
    }

    // Wave32 reduction of the per-lane partials.
#pragma unroll
    for (int off = 16; off > 0; off >>= 1)
        acc += __shfl_xor(acc, off, 32);

    if (lane == 0)
        per_b[b] = acc * (1.0f / (float)C_CTX);   // = oloss[b] + nloss[b]
}

__global__ __launch_bounds__(256) void reduce_mean_neg_kernel(
    const float* __restrict__ per_b, float* __restrict__ out, int Bn)
{
    __shared__ float sm[256];
    float s = 0.0f;
    for (int i = threadIdx.x; i < Bn; i += 256) s += per_b[i];
    sm[threadIdx.x] = s;
    __syncthreads();
    for (int off = 128; off > 0; off >>= 1) {
        if ((int)threadIdx.x < off) sm[threadIdx.x] += sm[threadIdx.x + off];
        __syncthreads();
    }
    if (threadIdx.x == 0) out[0] = -sm[0] / (float)Bn;
}

extern "C" void kernel_launch(void* const* d_in, const int* in_sizes, int n_in,
                              void* d_out, int out_size, void* d_ws, size_t ws_size,
                              hipStream_t stream) {
    const int*   iword  = (const int*)d_in[0];
    const int*   owords = (const int*)d_in[1];
    const int*   nwords = (const int*)d_in[2];
    const float* emb1   = (const float*)d_in[3];
    const float* emb2   = (const float*)d_in[4];
    float* per_b = (float*)d_ws;     // Bn floats of scratch
    float* out   = (float*)d_out;    // single f32 scalar

    const int Bn = in_sizes[0];
    const int blocks = (Bn + 7) / 8; // 8 waves (one batch element each) per block

    sgns_wmma_kernel<<<blocks, 256, 0, stream>>>(iword, owords, nwords,
                                                 emb1, emb2, per_b, Bn);
    reduce_mean_neg_kernel<<<1, 256, 0, stream>>>(per_b, out, Bn);
}